// DenseContrastive_41248865911089
// MI455X (gfx1250) — compile-verified
//
#include <hip/hip_runtime.h>

// DenseContrastive loss for MI455X (gfx1250, wave32).
// Streaming fp32 GEMM (A @ E^T, 12800x12800x64) fused with online row-wise
// logsumexp using V_WMMA_F32_16X16X4_F32.
//  - 8 column tiles per chunk, double-buffered in LDS (64 KB), staged with
//    GLOBAL_LOAD_ASYNC_TO_LDS_B128 when available (ASYNCcnt), else ld+ds_store.
//  - K-loop interleaves 8 independent accumulator chains to hide WMMA latency.
//  - Softmax state merged per chunk: 1 rescale + 8 exp2 per row.
//  - pos column folded at the end; deterministic 2-kernel mean reduction.

typedef __attribute__((ext_vector_type(2))) float v2f;
typedef __attribute__((ext_vector_type(8))) float v8f;
typedef __attribute__((ext_vector_type(4))) int   v4i;

typedef __attribute__((address_space(1))) v4i* gv4i_ptr;  // global int4*
typedef __attribute__((address_space(3))) v4i* lv4i_ptr;  // LDS int4*

#define HW            6400      // 80*80 anchors per batch
#define CNL           64        // channels
#define BATCH_STRIDE  409600    // CNL*HW floats per batch
#define NTOT          12800     // 2*HW total anchors
#define ROWTILES      800       // NTOT/16
#define WAVES_PER_BLK 8
#define TILES_PER_CHK 8         // column tiles staged per chunk
#define NCHUNK        100       // 800/8
#define CHUNK_FLOATS  (TILES_PER_CHK * CNL * 16)   // 8192 floats = 32KB
#define SCALE_Z       14.4269504088896340736f      // (1/TEMP) * log2(e)
#define EPSF          1e-8f

#if __has_builtin(__builtin_amdgcn_global_load_async_to_lds_b128) && \
    __has_builtin(__builtin_amdgcn_s_wait_asynccnt)
#define USE_ASYNC_LDS 1
#else
#define USE_ASYNC_LDS 0
#endif

__global__ __launch_bounds__(256) void dc_rowtile_kernel(
    const float* __restrict__ Amain,   // proj_main (b,c,H,W)
    const float* __restrict__ Eema,    // proj_ema  (b,c,H,W)
    float* __restrict__ partial)       // per-rowtile loss sums (800)
{
  __shared__ __align__(16) float lds[2 * CHUNK_FLOATS];   // 64 KB, dbl-buffered

  const int tid  = threadIdx.x;
  const int wave = tid >> 5;
  const int lane = tid & 31;
  const int half = lane >> 4;   // which 16-lane half of the wave
  const int l    = lane & 15;

  const int rt   = blockIdx.x * WAVES_PER_BLK + wave;  // row tile 0..799
  const int n0   = rt * 16;
  const int bi   = (n0 >= HW) ? 1 : 0;                 // batch (tiles never cross)
  const int base = bi * BATCH_STRIDE + (n0 - bi * HW);

  // ---- Load A fragments in WMMA 16x4 layout; fold pos dot along the way.
  // Frag kt covers K = 4kt..4kt+3: lanes 0-15 hold (M=l, K0/K1), lanes 16-31
  // hold (M=l, K2/K3). Memory is K-major, n contiguous -> lane-contiguous runs.
  v2f a[16];
  float posP = 0.f;
  #pragma unroll
  for (int kt = 0; kt < 16; ++kt) {
    const int k0 = 4 * kt + 2 * half;
    const float ax = Amain[base + (k0    ) * HW + l];
    const float ay = Amain[base + (k0 + 1) * HW + l];
    const float ex = Eema [base + (k0    ) * HW + l];
    const float ey = Eema [base + (k0 + 1) * HW + l];
    a[kt].x = ax; a[kt].y = ay;
    posP += ax * ex + ay * ey;
  }
  posP += __shfl_xor(posP, 16);          // combine K halves: full pos dot
  const float zP = posP * SCALE_Z;       // pos logit (log2 units), row n0+(lane&15)

  // ---- Online softmax state: 8 rows per lane (row = 8*half + i).
  float m[8], s[8];
  #pragma unroll
  for (int i = 0; i < 8; ++i) { m[i] = -1e38f; s[i] = 0.f; }

  // ---- Cooperative staging of E column tiles into LDS (raw (t,k,m) layout).
  auto stage = [&](int buf, int cc) {
    #pragma unroll
    for (int j = 0; j < 8; ++j) {
      const int idx  = tid + j * 256;        // 0..2047 float4s per chunk
      const int quad = idx & 3;
      const int k    = (idx >> 2) & 63;
      const int t    = idx >> 8;             // 0..7
      const int cn0  = (cc * TILES_PER_CHK + t) * 16;
      const int cbi  = (cn0 >= HW) ? 1 : 0;
      const int ga   = cbi * BATCH_STRIDE + k * HW + (cn0 - cbi * HW) + quad * 4;
      const int lo   = buf * CHUNK_FLOATS + (t * CNL + k) * 16 + quad * 4;
#if USE_ASYNC_LDS
      __builtin_amdgcn_global_load_async_to_lds_b128(
          (gv4i_ptr)(Eema + ga), (lv4i_ptr)&lds[lo], 0, 0);
#else
      *(float4*)&lds[lo] = *(const float4*)(Eema + ga);
#endif
    }
  };

  int cur = 0;
  stage(0, 0);
  for (int cc = 0; cc < NCHUNK; ++cc) {
#if USE_ASYNC_LDS
    __builtin_amdgcn_s_wait_asynccnt(0);      // my async writes to LDS landed
#endif
    __syncthreads();                          // everyone's staging of `cur` done
    if (cc + 1 < NCHUNK) stage(1 - cur, cc + 1);
    const float* B = &lds[cur * CHUNK_FLOATS];

    // 8 independent accumulator chains, interleaved per K-step.
    v8f acc[TILES_PER_CHK];
    #pragma unroll
    for (int t = 0; t < TILES_PER_CHK; ++t)
      acc[t] = (v8f){0.f, 0.f, 0.f, 0.f, 0.f, 0.f, 0.f, 0.f};

    #pragma unroll
    for (int kt = 0; kt < 16; ++kt) {
      const int k0 = 4 * kt + 2 * half;       // B-frag: (N=l, K0/K1 | K2/K3)
      v2f bf[TILES_PER_CHK];
      #pragma unroll
      for (int t = 0; t < TILES_PER_CHK; ++t) {
        const float* Bt = B + t * (CNL * 16);
        bf[t].x = Bt[(k0    ) * 16 + l];
        bf[t].y = Bt[(k0 + 1) * 16 + l];
      }
      #pragma unroll
      for (int t = 0; t < TILES_PER_CHK; ++t)
        acc[t] = __builtin_amdgcn_wmma_f32_16x16x4_f32(
            false, a[kt], false, bf[t], (short)0, acc[t], false, false);
    }

    // Merged online logsumexp update: one max + one rescale + 8 exp2 per row.
    #pragma unroll
    for (int i = 0; i < 8; ++i) {
      float z[TILES_PER_CHK];
      #pragma unroll
      for (int t = 0; t < TILES_PER_CHK; ++t) z[t] = acc[t][i] * SCALE_Z;
      float zm = z[0];
      #pragma unroll
      for (int t = 1; t < TILES_PER_CHK; ++t) zm = fmaxf(zm, z[t]);
      const float nm = fmaxf(m[i], zm);
      float add = 0.f;
      #pragma unroll
      for (int t = 0; t < TILES_PER_CHK; ++t) add += exp2f(z[t] - nm);
      s[i] = s[i] * exp2f(m[i] - nm) + add;
      m[i] = nm;
    }
    cur = 1 - cur;
  }

  // ---- Butterfly-combine softmax state across the 16 lanes of each half.
  #pragma unroll
  for (int off = 1; off <= 8; off <<= 1) {
    #pragma unroll
    for (int i = 0; i < 8; ++i) {
      const float mo = __shfl_xor(m[i], off);
      const float so = __shfl_xor(s[i], off);
      const float nm = fmaxf(m[i], mo);
      s[i] = s[i] * exp2f(m[i] - nm) + so * exp2f(mo - nm);
      m[i] = nm;
    }
  }

  // ---- Fold pos column, compute per-row loss, wave-sum, write partial.
  float myloss = 0.f;
  #pragma unroll
  for (int i = 0; i < 8; ++i) {
    const int   r  = 8 * half + i;           // row within tile, 0..15
    const float pz = __shfl(zP, r);          // pos logit lives in lane r
    const float nm = fmaxf(m[i], pz);
    const float S  = s[i] * exp2f(m[i] - nm) + exp2f(pz - nm);  // denom incl. pos
    const float p  = exp2f(pz - nm);
    const float Lr = -logf(p / (S + EPSF) + EPSF);
    if (l == i) myloss = Lr;                 // lanes 0..7 of each half keep one row
  }
  float wsum = myloss;
  #pragma unroll
  for (int off = 1; off < 32; off <<= 1) wsum += __shfl_xor(wsum, off);
  if (lane == 0) partial[rt] = wsum;         // sum of this tile's 16 row losses
}

__global__ __launch_bounds__(256) void dc_reduce_kernel(
    const float* __restrict__ partial, float* __restrict__ out)
{
  __shared__ float sm[256];
  const int t = threadIdx.x;
  float a = 0.f;
  for (int i = t; i < ROWTILES; i += 256) a += partial[i];
  sm[t] = a;
  __syncthreads();
  for (int sft = 128; sft > 0; sft >>= 1) {
    if (t < sft) sm[t] += sm[t + sft];
    __syncthreads();
  }
  if (t == 0) out[0] = sm[0] / (float)NTOT;
}

extern "C" void kernel_launch(void* const* d_in, const int* in_sizes, int n_in,
                              void* d_out, int out_size, void* d_ws, size_t ws_size,
                              hipStream_t stream) {
  (void)in_sizes; (void)n_in; (void)out_size; (void)ws_size;
  const float* proj_main = (const float*)d_in[0];
  const float* proj_ema  = (const float*)d_in[1];
  // d_in[2] (label_main), d_in[3] (label_ema), d_in[4] (patch_num): unused by the loss.
  float* ws  = (float*)d_ws;    // 800 partial sums
  float* out = (float*)d_out;

  dc_rowtile_kernel<<<ROWTILES / WAVES_PER_BLK, 256, 0, stream>>>(proj_main, proj_ema, ws);
  dc_reduce_kernel<<<1, 256, 0, stream>>>(ws, out);
}